// ECCONetwork_44023414784209
// MI455X (gfx1250) — compile-verified
//
#include <hip/hip_runtime.h>
#include <hip/hip_bf16.h>
#include <math.h>

// ---------------- problem constants (match reference) ----------------
#define NRAD   3
#define NTH    16
#define NBINS  48            // NRAD*NTH
#define RADIUS_F 40.0f
#define DT_F     0.1f
#define SHIFT_F  0.2f
#define BB     4
#define NN     512
#define NBOX   512
#define PI_F   3.14159265358979323846f
#define CHUNK  32            // K-rows staged per async LDS chunk

typedef __attribute__((ext_vector_type(2))) float v2f;
typedef __attribute__((ext_vector_type(8))) float v8f;

// ---------------------------------------------------------------------
// Kernel 1: integrate v1/p1 and assemble fluid_feats = [v1, other(3), v0_enc(16)]
// feats layout: [B*N, 20, 2]
// ---------------------------------------------------------------------
__global__ void k_prep(const float* __restrict__ p0, const float* __restrict__ v0,
                       const float* __restrict__ acc, const float* __restrict__ other,
                       const float* __restrict__ v0e,
                       float* __restrict__ p1, float* __restrict__ feats)
{
    int idx = blockIdx.x * blockDim.x + threadIdx.x;
    if (idx >= BB * NN) return;
    float v0x = v0[idx*2+0], v0y = v0[idx*2+1];
    float v1x = v0x + DT_F * acc[idx*2+0];
    float v1y = v0y + DT_F * acc[idx*2+1];
    p1[idx*2+0] = p0[idx*2+0] + DT_F * (v0x + v1x) * 0.5f;
    p1[idx*2+1] = p0[idx*2+1] + DT_F * (v0y + v1y) * 0.5f;
    float* f = feats + (size_t)idx * 40;
    f[0] = v1x; f[1] = v1y;
#pragma unroll
    for (int j = 0; j < 6; ++j)  f[2+j] = other[(size_t)idx*6 + j];
#pragma unroll
    for (int j = 0; j < 32; ++j) f[8+j] = v0e[(size_t)idx*32 + j];
}

// ---------------------------------------------------------------------
// Kernel 2: build combined GEMM kernel matrix K_buf[KDpad x ncols].
// rows (k<48,i,v): steered conv kernel R(t) W R(t)^T  -> cols [2*conv_o0, ...)
// rows (k==48,i,v): equi-linear (a*I + b*J)           -> cols [2*dense_o0, ...)
// everything else zero (incl. pad rows/cols).
// ---------------------------------------------------------------------
__global__ void k_build_K(const float* __restrict__ W, const float* __restrict__ A,
                          const float* __restrict__ Bm, float* __restrict__ Kb,
                          int I, int O_conv, int conv_o0, int O_dense, int dense_o0,
                          int ncols, int KDpad)
{
    int idx = blockIdx.x * blockDim.x + threadIdx.x;
    if (idx >= KDpad * ncols) return;
    int row = idx / ncols, col = idx - row * ncols;
    int i2 = I * 2;
    int kbin = row / i2, rem = row - kbin * i2;
    int i = rem >> 1, v = rem & 1;
    int op = col >> 1, u = col & 1;
    float val = 0.f;
    if (kbin < NBINS) {
        int o = op - conv_o0;
        if (o >= 0 && o < O_conv) {
            int rb = kbin >> 4, t = kbin & 15;
            float th = ((float)t + 0.5f) * (2.f * PI_F / (float)NTH) - PI_F;
            float c = __cosf(th), s = __sinf(th);
            const float* Wp = W + (size_t)(((rb*NTH + t)*O_conv + o)*I + i) * 4;
            float w00 = Wp[0], w01 = Wp[1], w10 = Wp[2], w11 = Wp[3];
            // K[u][v] = sum_{p,q} R[u][p] * W[p][q] * R[v][q],  R = [[c,-s],[s,c]]
            float ru0 = u ? s : c, ru1 = u ? c : -s;
            float rv0 = v ? s : c, rv1 = v ? c : -s;
            val = ru0 * (w00*rv0 + w01*rv1) + ru1 * (w10*rv0 + w11*rv1);
        }
    } else if (kbin == NBINS && A != nullptr) {
        int o = op - dense_o0;
        if (o >= 0 && o < O_dense) {
            float a = A[o*I + i], b = Bm[o*I + i];
            // od[o,0] = a*f0 - b*f1 ; od[o,1] = b*f0 + a*f1
            val = (u == 0) ? (v == 0 ?  a : -b)
                           : (v == 0 ?  b :  a);
        }
    }
    Kb[idx] = val;
}

// ---------------------------------------------------------------------
// Kernel 3: polar-bin pooling. One workgroup per (b, q).
// LDS accumulator Mloc[48*I*2] via ds_add_f32; then the padded M row
// (bins | query-feature tail | zero pad) is streamed to global scratch.
// ---------------------------------------------------------------------
__global__ __launch_bounds__(256)
void k_bin_accum(const float* __restrict__ src, const float* __restrict__ qry,
                 const float* __restrict__ feats, const float* __restrict__ mask,
                 float* __restrict__ Mg, int S, int I, int KDpad)
{
    __shared__ float Mloc[NBINS * 48 * 2];      // max I = 48 -> 18 KB
    const int bq = blockIdx.x;
    const int b  = bq / NN, q = bq - b * NN;
    const int KBIN = NBINS * I * 2;

    for (int e = threadIdx.x; e < KBIN; e += blockDim.x) Mloc[e] = 0.f;
    __syncthreads();

    const float qx = qry[(size_t)bq*2+0], qy = qry[(size_t)bq*2+1];
    const float* mrow = mask + (size_t)bq * S;
    for (int s = threadIdx.x; s < S; s += blockDim.x) {
        float rx = src[(size_t)(b*S + s)*2+0] - qx;
        float ry = src[(size_t)(b*S + s)*2+1] - qy;
        float d2 = rx*rx + ry*ry;
        float win = fmaxf(1.f - d2 * (1.f/(RADIUS_F*RADIUS_F)), 0.f);
        float w = win * win * win * mrow[s];
        if (w != 0.f) {
            float dist = sqrtf(d2 + 1e-9f);
            int rb = (int)(dist * ((float)NRAD / RADIUS_F));
            rb = rb > (NRAD-1) ? (NRAD-1) : (rb < 0 ? 0 : rb);
            float th = atan2f(ry, rx);
            int tb = (int)floorf((th + PI_F) * ((float)NTH / (2.f*PI_F)));
            tb &= (NTH - 1);
            int base = (rb*NTH + tb) * I * 2;
            const float* fp = feats + (size_t)(b*S + s) * I * 2;
            for (int j = 0; j < I*2; ++j)
                atomicAdd(&Mloc[base + j], w * fp[j]);
        }
    }
    __syncthreads();

    float* Mrow = Mg + (size_t)bq * KDpad;
    const float* fq = feats + (size_t)(b*S + q) * I * 2;   // dense tail (query feats)
    const int KD = (NBINS + 1) * I * 2;
    for (int e = threadIdx.x; e < KDpad; e += blockDim.x) {
        float v;
        if (e < KBIN)      v = Mloc[e];
        else if (e < KD)   v = fq[e - KBIN];
        else               v = 0.f;
        Mrow[e] = v;
    }
}

// ---------------------------------------------------------------------
// Kernel 4: fp32 WMMA GEMM with async-LDS staged B panel.
//   OUT[2048 x ncols] = M[2048 x KDpad] * K[KDpad x ncols]
// 8 waves/block; wave w owns row tile blockIdx.x*128 + 16w; all waves share
// the 16-column B panel, staged chunk-by-chunk (32 k-rows = 2 KB) into LDS
// with GLOBAL_LOAD_ASYNC_TO_LDS_B64 (ASYNCcnt), double-buffered so the DMA
// of chunk c+1 overlaps the 8 V_WMMA_F32_16X16X4_F32 steps of chunk c.
// f32 A layout: lanes 0-15 hold (M=lane, K=kb+{0,1}); lanes 16-31 K=kb+{2,3}.
// C/D layout: VGPR r -> M = r + 8*(lane>=16), N = lane&15.
// ---------------------------------------------------------------------
__global__ __launch_bounds__(256)
void k_gemm(const float* __restrict__ Mg, const float* __restrict__ Kb,
            float* __restrict__ Out, int KDpad, int ncols,
            int ch_base, int CH, int O_valid, int residual)
{
    __shared__ __align__(16) float Bsh[2][CHUNK * 16];
    const int tid  = threadIdx.x;
    const int wave = tid >> 5;
    const int lane = tid & 31;
    const int lrow = lane & 15;
    const int half = lane >> 4;
    const int row0 = blockIdx.x * 128 + wave * 16;
    const int col0 = blockIdx.y * 16;

    const float* Arow = Mg + (size_t)(row0 + lrow) * KDpad + 2 * half;

    // this thread stages float-pair e0 = 2*tid of each 512-float chunk
    const int e0    = tid * 2;          // even -> pair stays within one k-row
    const int st_kr = e0 >> 4;          // chunk-local k row
    const int st_c  = e0 & 15;          // column within 16-wide tile

    {   // stage chunk 0 -> buffer 0
        const float* g = Kb + (size_t)st_kr * ncols + col0 + st_c;
        unsigned l = (unsigned)(size_t)&Bsh[0][e0];
        asm volatile("global_load_async_to_lds_b64 %0, %1, off"
                     :: "v"(l), "v"(g) : "memory");
    }

    v8f acc = {0.f,0.f,0.f,0.f,0.f,0.f,0.f,0.f};
    const int nchunk = KDpad / CHUNK;
    for (int c = 0; c < nchunk; ++c) {
        asm volatile("s_wait_asynccnt 0x0" ::: "memory");
        __syncthreads();                                   // Bsh[c&1] visible to all
        if (c + 1 < nchunk) {                              // restage other buffer
            const float* g = Kb + (size_t)((c + 1) * CHUNK + st_kr) * ncols
                                + col0 + st_c;
            unsigned l = (unsigned)(size_t)&Bsh[(c + 1) & 1][e0];
            asm volatile("global_load_async_to_lds_b64 %0, %1, off"
                         :: "v"(l), "v"(g) : "memory");
        }
        const float* As = Arow + c * CHUNK;
        const float* Bs = &Bsh[c & 1][0];
        __builtin_prefetch(As + CHUNK, 0, 3);              // global_prefetch next A chunk
#pragma unroll
        for (int kk = 0; kk < CHUNK; kk += 4) {
            v2f a, b;
            a.x = As[kk + 0];
            a.y = As[kk + 1];
            int kr = kk + 2 * half;
            b.x = Bs[kr * 16 + lrow];
            b.y = Bs[(kr + 1) * 16 + lrow];
            acc = __builtin_amdgcn_wmma_f32_16x16x4_f32(
                      false, a, false, b, (short)0, acc, false, false);
        }
        __syncthreads();                                   // done reading before restage
    }

    const int cg = col0 + lrow;
    const int op = cg >> 1, u = cg & 1;
    if (op < O_valid) {
#pragma unroll
        for (int r = 0; r < 8; ++r) {
            int m = r + 8 * half;
            size_t idx = ((size_t)(row0 + m) * CH + ch_base + op) * 2 + u;
            float v = acc[r];
            if (residual) v += Out[idx];
            Out[idx] = v;
        }
    }
}

// ---------------------------------------------------------------------
// Kernel 5: shifted-magnitude nonlinearity: f = out/mags * relu(mags - 0.2)
// ---------------------------------------------------------------------
__global__ void k_nonlin(const float* __restrict__ Outb, float* __restrict__ feats, int total)
{
    int idx = blockIdx.x * blockDim.x + threadIdx.x;
    if (idx >= total) return;
    float x = Outb[idx*2+0], y = Outb[idx*2+1];
    float mags = x*x + y*y + 1e-6f;
    float sc = fmaxf(mags - SHIFT_F, 0.f) / mags;
    feats[idx*2+0] = x * sc;
    feats[idx*2+1] = y * sc;
}

// ---------------------------------------------------------------------
// Kernel 6: apply correction, emit (p_corrected, v_corrected)
// ---------------------------------------------------------------------
__global__ void k_final(const float* __restrict__ Outb, const float* __restrict__ p1,
                        const float* __restrict__ p0, float* __restrict__ out, int n)
{
    int idx = blockIdx.x * blockDim.x + threadIdx.x;
    if (idx >= n) return;
    float cx = Outb[idx*2+0] * (1.f/128.f);
    float cy = Outb[idx*2+1] * (1.f/128.f);
    float px = p1[idx*2+0] + cx;
    float py = p1[idx*2+1] + cy;
    out[idx*2+0] = px;
    out[idx*2+1] = py;
    out[(size_t)n*2 + idx*2+0] = (px - p0[idx*2+0]) * (1.f/DT_F);
    out[(size_t)n*2 + idx*2+1] = (py - p0[idx*2+1]) * (1.f/DT_F);
}

// ---------------------------------------------------------------------
extern "C" void kernel_launch(void* const* d_in, const int* in_sizes, int n_in,
                              void* d_out, int out_size, void* d_ws, size_t ws_size,
                              hipStream_t stream)
{
    (void)in_sizes; (void)n_in; (void)out_size; (void)ws_size;
    const float* v0_enc = (const float*)d_in[1];
    const float* p0     = (const float*)d_in[2];
    const float* v0     = (const float*)d_in[3];
    const float* acc    = (const float*)d_in[4];
    const float* other  = (const float*)d_in[5];
    const float* box    = (const float*)d_in[6];
    const float* boxf   = (const float*)d_in[7];
    const float* fmask  = (const float*)d_in[8];
    const float* bmask  = (const float*)d_in[9];
    const float* W_cf   = (const float*)d_in[10];
    const float* W_co   = (const float*)d_in[11];
    const float* A_df   = (const float*)d_in[12];
    const float* B_df   = (const float*)d_in[13];
    const float* W_c[4] = {(const float*)d_in[14], (const float*)d_in[17],
                           (const float*)d_in[20], (const float*)d_in[23]};
    const float* A_d[4] = {(const float*)d_in[15], (const float*)d_in[18],
                           (const float*)d_in[21], (const float*)d_in[24]};
    const float* B_d[4] = {(const float*)d_in[16], (const float*)d_in[19],
                           (const float*)d_in[22], (const float*)d_in[25]};

    // workspace layout (floats)
    float* ws    = (float*)d_ws;
    float* p1    = ws;                                 // 4096
    float* feats = p1    + BB*NN*2;                    // 2048*48*2 = 196608
    float* outb  = feats + (size_t)BB*NN*48*2;         // 196608
    float* Kb    = outb  + (size_t)BB*NN*48*2;         // 4704*64 = 301056
    float* Mg    = Kb    + (size_t)4704*64;            // 2048*4704 = 9633792

    const int BN = BB * NN;                            // 2048 rows

    k_prep<<<(BN + 255)/256, 256, 0, stream>>>(p0, v0, acc, other, v0_enc, p1, feats);

    // ---- stage A: fluid conv (ch 16..31) + dense (ch 32..47), I=20 ----
    {
        const int I = 20, KDpad = 1984 /*49*20*2=1960 -> pad to 32x*/, ncols = 64;
        k_build_K<<<(KDpad*ncols + 255)/256, 256, 0, stream>>>(
            W_cf, A_df, B_df, Kb, I, 16, 0, 16, 16, ncols, KDpad);
        k_bin_accum<<<BN, 256, 0, stream>>>(p1, p1, feats, fmask, Mg, NN, I, KDpad);
        k_gemm<<<dim3(BN/128, ncols/16), 256, 0, stream>>>(
            Mg, Kb, outb, KDpad, ncols, /*ch_base=*/16, /*CH=*/48, /*O_valid=*/32, 0);
    }
    // ---- stage B: obstacle conv (ch 0..15), I=1 ----
    {
        const int I = 1, KDpad = 128 /*98 -> pad to 32x*/, ncols = 32;
        k_build_K<<<(KDpad*ncols + 255)/256, 256, 0, stream>>>(
            W_co, nullptr, nullptr, Kb, I, 16, 0, 0, 0, ncols, KDpad);
        k_bin_accum<<<BN, 256, 0, stream>>>(box, p1, boxf, bmask, Mg, NBOX, I, KDpad);
        k_gemm<<<dim3(BN/128, ncols/16), 256, 0, stream>>>(
            Mg, Kb, outb, KDpad, ncols, /*ch_base=*/0, /*CH=*/48, /*O_valid=*/16, 0);
    }
    // ---- 4 residual layers: (O,I) = (32,48)(32,32)(32,32)(1,32) ----
    const int Os[4] = {32, 32, 32, 1};
    const int Is[4] = {48, 32, 32, 32};
    for (int l = 0; l < 4; ++l) {
        const int I = Is[l], O = Os[l];
        const int KDpad = (NBINS+1)*I*2;               // 4704 / 3136: multiples of 32
        const int ncols = ((2*O + 15)/16)*16;
        k_nonlin<<<(BN*I + 255)/256, 256, 0, stream>>>(outb, feats, BN*I);
        k_build_K<<<(KDpad*ncols + 255)/256, 256, 0, stream>>>(
            W_c[l], A_d[l], B_d[l], Kb, I, O, 0, O, 0, ncols, KDpad);
        k_bin_accum<<<BN, 256, 0, stream>>>(p1, p1, feats, fmask, Mg, NN, I, KDpad);
        const int residual = (l == 1 || l == 2) ? 1 : 0;
        k_gemm<<<dim3(BN/128, ncols/16), 256, 0, stream>>>(
            Mg, Kb, outb, KDpad, ncols, /*ch_base=*/0, /*CH=*/O, /*O_valid=*/O, residual);
    }

    k_final<<<(BN + 255)/256, 256, 0, stream>>>(outb, p1, p0, (float*)d_out, BN);
}